// HierarchicalCrossEntropyLoss_8770323219049
// MI455X (gfx1250) — compile-verified
//
#include <hip/hip_runtime.h>
#include <hip/hip_bf16.h>

// ---------------------------------------------------------------------------
// Hierarchical (wordtree [32,32]) cross-entropy loss, MI455X / gfx1250.
// Per sample: 1-2 contiguous 32-float sibling groups (each exactly one 128B
// cacheline), stable log-softmax over 32 lanes of a wave32, mean over N.
// CDNA5 paths used: global_load_async_to_lds_b32 double-buffer pipeline
// (ASYNCcnt), wave32 ds_swizzle XOR reductions, ds_bpermute slot pick.
// ---------------------------------------------------------------------------

#define SWZ_XOR_F(x, xmask) \
  __int_as_float(__builtin_amdgcn_ds_swizzle(__float_as_int(x), (((xmask) << 10) | 0x1f)))

// Async copy of one dword/lane from global into LDS (CDNA5 async path,
// tracked by ASYNCcnt). LDS byte address = low 32 bits of the flat address.
__device__ __forceinline__ void async_copy_b32(const float* g, const float* lds) {
  unsigned lds_addr = (unsigned)(unsigned long long)lds;
  unsigned long long gaddr = (unsigned long long)g;
  asm volatile("global_load_async_to_lds_b32 %0, %1, off"
               :: "v"(lds_addr), "v"(gaddr) : "memory");
}

// 32-wide stable logsumexp over a group staged in LDS; returns NLL of `pos`.
// All values are wave-uniform on return. pos is wave-uniform.
__device__ __forceinline__ float wave_group_nll(const float* slot, int lane, int pos) {
  float v = slot[lane];
  float m = v;
  m = fmaxf(m, SWZ_XOR_F(m, 16));
  m = fmaxf(m, SWZ_XOR_F(m, 8));
  m = fmaxf(m, SWZ_XOR_F(m, 4));
  m = fmaxf(m, SWZ_XOR_F(m, 2));
  m = fmaxf(m, SWZ_XOR_F(m, 1));
  float e = __expf(v - m);
  float s = e;
  s += SWZ_XOR_F(s, 16);
  s += SWZ_XOR_F(s, 8);
  s += SWZ_XOR_F(s, 4);
  s += SWZ_XOR_F(s, 2);
  s += SWZ_XOR_F(s, 1);
  float vp = __int_as_float(__builtin_amdgcn_ds_bpermute(pos << 2, __float_as_int(v)));
  return m + __logf(s) - vp;   // = -(log_softmax[pos])
}

__global__ __launch_bounds__(256) void hce_main_kernel(
    const float* __restrict__ cls, const int* __restrict__ label,
    float* __restrict__ partial, long long N, int C, int spw) {
  // [wave][pingpong][group(level0/level1)][lane]
  __shared__ float buf[8][2][2][32];
  __shared__ float wsum[8];

  const int lane = (int)threadIdx.x & 31;
  const int w    = (int)threadIdx.x >> 5;
  const long long wid = (long long)blockIdx.x * 8 + w;

  long long s0   = wid * (long long)spw;
  long long sEnd = s0 + spw;
  if (sEnd > N) sEnd = N;

  // Issue the two 128B sibling-group lines of sample `s` into buffer `pp`.
  // Returns the (wave-uniform) label.
  auto issue = [&](long long s, int pp) -> int {
    int l = label[s];                       // wave-uniform -> scalar load
    int base0 = 0;                          // level-0 group base column
    if (l > 32) { int t = l - 33; base0 = 32 + ((t >> 5) << 5); }
    const float* row = cls + s * (long long)C;
    // ensure prior ds reads of the buffer being overwritten have drained
    asm volatile("s_wait_dscnt 0x0" ::: "memory");
    async_copy_b32(row + base0 + lane, &buf[w][pp][0][lane]);  // deepest group
    async_copy_b32(row + lane,         &buf[w][pp][1][lane]);  // level-1 group
    return l;
  };

  float acc = 0.f;
  if (s0 < sEnd) {
    int pp = 0;
    int lcur = issue(s0, 0);
    for (long long s = s0; s < sEnd; ++s, pp ^= 1) {
      int lnext = 0;
      if (s + 1 < sEnd) {
        lnext = issue(s + 1, pp ^ 1);
        // 2 async ops still in flight (next sample); current batch has landed
        asm volatile("s_wait_asynccnt 0x2" ::: "memory");
      } else {
        asm volatile("s_wait_asynccnt 0x0" ::: "memory");
      }
      // Wave-uniform control flow: lcur identical across the wave.
      if (lcur > 0) {
        int pos0, p = -1;
        if (lcur <= 32) {
          pos0 = lcur - 1;
        } else {
          int t = lcur - 33;
          p = t >> 5;          // parent-1 == position within level-1 group
          pos0 = t & 31;
        }
        acc += wave_group_nll(&buf[w][pp][0][0], lane, pos0);
        if (p >= 0)
          acc += wave_group_nll(&buf[w][pp][1][0], lane, p);
      }
      lcur = lnext;
    }
  }

  // acc is wave-uniform; reduce 8 waves -> one partial per block.
  if (lane == 0) wsum[w] = acc;
  __syncthreads();
  if (threadIdx.x == 0) {
    float t = 0.f;
#pragma unroll
    for (int i = 0; i < 8; ++i) t += wsum[i];
    partial[blockIdx.x] = t;
  }
}

__global__ __launch_bounds__(256) void hce_finalize_kernel(
    const float* __restrict__ partial, int n, float scale, float* __restrict__ out) {
  __shared__ float s[256];
  float a = 0.f;
  for (int i = (int)threadIdx.x; i < n; i += 256) a += partial[i];
  s[threadIdx.x] = a;
  __syncthreads();
  for (int o = 128; o > 0; o >>= 1) {
    if ((int)threadIdx.x < o) s[threadIdx.x] += s[threadIdx.x + o];
    __syncthreads();
  }
  if (threadIdx.x == 0) out[0] = s[0] * scale;
}

extern "C" void kernel_launch(void* const* d_in, const int* in_sizes, int n_in,
                              void* d_out, int out_size, void* d_ws, size_t ws_size,
                              hipStream_t stream) {
  const float* cls  = (const float*)d_in[0];
  const int* label  = (const int*)d_in[1];
  long long N = (n_in > 1) ? (long long)in_sizes[1] : 0;
  if (N <= 0) N = 1;
  int C = (int)(in_sizes[0] / N);          // 1056 for the [32,32] tree

  const int THREADS = 256;                  // 8 wave32s per block
  int blocks = 1024;
  if ((size_t)blocks * sizeof(float) > ws_size)
    blocks = (int)(ws_size / sizeof(float));
  if (blocks < 1) blocks = 1;

  long long wavesTotal = (long long)blocks * (THREADS / 32);
  int spw = (int)((N + wavesTotal - 1) / wavesTotal);   // samples per wave

  float* partial = (float*)d_ws;
  hce_main_kernel<<<blocks, THREADS, 0, stream>>>(cls, label, partial, N, C, spw);
  hce_finalize_kernel<<<1, 256, 0, stream>>>(partial, blocks,
                                             (float)(1.0 / (double)N),
                                             (float*)d_out);
}